// LabChannelLoss_50483045597466
// MI455X (gfx1250) — compile-verified
//
#include <hip/hip_runtime.h>
#include <hip/hip_bf16.h>

typedef __attribute__((ext_vector_type(2))) float v2f;
typedef __attribute__((ext_vector_type(8))) float v8f;

#define N_IMG 16
#define HW 512
#define PLANE (HW * HW)          // 262144
#define NPIX (N_IMG * PLANE)     // 4194304
#define EPS_LAB 0.008856f

// Padded intermediate plane: 8-cell zero halo on every side.
#define PW 528
#define PPLANE (PW * PW)         // 278784
#define NPLANES (N_IMG * 3)      // 48
#define BORDER_PER_PLANE 16640   // 2*8*528 + 2*8*512
#define NBORDER (NPLANES * BORDER_PER_PLANE) // 798720

#define TILES (N_IMG * 3 * 32 * 32)     // 49152 16x16 tiles
#define WAVES_PER_BLOCK 4
#define NBLK2 (TILES / WAVES_PER_BLOCK) // 12288
#define LDS_STRIDE 17

__device__ __forceinline__ float f_lab(float t) {
    return (t > EPS_LAB) ? cbrtf(t) : (7.787f * t + 16.0f / 116.0f);
}

__device__ __forceinline__ void rgb2lab(float r, float g, float b,
                                        float& L, float& A, float& B) {
    r *= (1.0f / 255.0f); g *= (1.0f / 255.0f); b *= (1.0f / 255.0f);
    float x = 0.4124564f * r + 0.3575761f * g + 0.1804375f * b;
    float y = 0.2126729f * r + 0.7151522f * g + 0.0721750f * b;
    float z = 0.0193339f * r + 0.1191920f * g + 0.9503041f * b;
    x *= (1.0f / 0.95047f);
    z *= (1.0f / 1.08883f);
    float fx = f_lab(x), fy = f_lab(y), fz = f_lab(z);
    L = 116.0f * fy - 16.0f;
    A = 500.0f * (fx - fy);
    B = 200.0f * (fy - fz);
}

// Pass 0: zero the halo border of every padded plane (runs every call).
__global__ __launch_bounds__(256) void zeropad_kernel(float* __restrict__ out) {
    int tid = blockIdx.x * blockDim.x + threadIdx.x;
    if (tid >= NBORDER) return;
    int p = tid / BORDER_PER_PLANE;
    int r = tid % BORDER_PER_PLANE;
    int row, col;
    if (r < 8 * PW) {                       // top 8 rows
        row = r / PW; col = r % PW;
    } else if (r < 16 * PW) {               // bottom 8 rows
        int q = r - 8 * PW; row = 520 + q / PW; col = q % PW;
    } else {                                // side columns, rows 8..519
        int q = r - 16 * PW;
        row = 8 + (q >> 4);
        int c = q & 15;
        col = (c < 8) ? c : (512 + c);      // 0..7 and 520..527
    }
    out[(size_t)p * PPLANE + row * PW + col] = 0.0f;
}

// Pass 1: labdiff (padded layout) = lab(input) - lab(target).
__global__ __launch_bounds__(256) void labdiff_kernel(const float* __restrict__ in,
                                                      const float* __restrict__ tg,
                                                      float* __restrict__ out) {
    int tid = blockIdx.x * blockDim.x + threadIdx.x;
    int n = tid >> 18;              // / 262144
    int pix = tid & (PLANE - 1);
    int h = pix >> 9, w = pix & (HW - 1);
    size_t ibase = (size_t)n * 3 * PLANE + pix;
    float Li, Ai, Bi, Lt, At, Bt;
    rgb2lab(in[ibase], in[ibase + PLANE], in[ibase + 2 * PLANE], Li, Ai, Bi);
    rgb2lab(tg[ibase], tg[ibase + PLANE], tg[ibase + 2 * PLANE], Lt, At, Bt);
    size_t obase = (size_t)n * 3 * PPLANE + (size_t)(h + 8) * PW + (w + 8);
    out[obase]              = Li - Lt;
    out[obase + PPLANE]     = Ai - At;
    out[obase + 2 * PPLANE] = Bi - Bt;
}

// Pass 2: 15x15 box sum via banded-matrix WMMA (horizontal then vertical),
// square, and deterministic per-block partial reduction. All loads are
// unconditional thanks to the zero halo.
__global__ __launch_bounds__(128) void boxwmma_kernel(const float* __restrict__ d,
                                                      float* __restrict__ partial) {
    __shared__ float hbuf[WAVES_PER_BLOCK][32 * LDS_STRIDE];
    __shared__ float wsum[WAVES_PER_BLOCK];

    const int lane = threadIdx.x & 31;
    const int wave = threadIdx.x >> 5;
    const int tile = blockIdx.x * WAVES_PER_BLOCK + wave;

    const int tc = tile & 31;
    const int tr = (tile >> 5) & 31;
    const int c  = (tile >> 10) % 3;
    const int n  = tile / 3072;
    const int r0 = tr * 16, c0 = tc * 16;

    const float* plane = d + (size_t)(n * 3 + c) * PPLANE;

    const int m  = lane & 15;   // A-frag: M index; B/D-frag: N index
    const int hi = lane >> 4;   // upper half-wave holds upper K / upper M

    // ---- Horizontal pass: H(32x16) = D(rows x 30 cols) * BandH(30x16) ----
    for (int g = 0; g < 2; ++g) {
        v8f acc = {};
        // padded row = 8 + (r0-7+16g+m); padded col base = 8 + (c0-7+2hi)
        const float* rowp = plane + (size_t)(r0 + 1 + 16 * g + m) * PW
                                  + (c0 + 1 + 2 * hi);
#pragma unroll
        for (int j = 0; j < 8; ++j) {
            const int ka = 4 * j + 2 * hi;       // absolute K (0..31)
            v2f a, b;
            a.x = rowp[4 * j];
            a.y = rowp[4 * j + 1];
            // band: output col n=m gathers input cols with n <= k <= n+14
            b.x = ((unsigned)(ka - m)     <= 14u) ? 1.0f : 0.0f;
            b.y = ((unsigned)(ka + 1 - m) <= 14u) ? 1.0f : 0.0f;
            acc = __builtin_amdgcn_wmma_f32_16x16x4_f32(
                false, a, false, b, (short)0, acc, false, false);
        }
        // D-frag element v: (row = v + 8*hi, col = m) -> LDS as H[32][16]
#pragma unroll
        for (int v = 0; v < 8; ++v)
            hbuf[wave][(16 * g + v + 8 * hi) * LDS_STRIDE + m] = acc[v];
    }

    __syncthreads();   // order LDS stores before loads (uniform across block)

    // ---- Vertical pass: Out(16x16) = BandV(16x32) * H(32x16) ----
    v8f outf = {};
#pragma unroll
    for (int j = 0; j < 8; ++j) {
        const int ka = 4 * j + 2 * hi;
        v2f a, b;
        // band: output row m gathers H rows with m <= k <= m+14
        a.x = ((unsigned)(ka - m)     <= 14u) ? 1.0f : 0.0f;
        a.y = ((unsigned)(ka + 1 - m) <= 14u) ? 1.0f : 0.0f;
        // B-frag: lane gives N=m, element gives K row of H
        b.x = hbuf[wave][ka * LDS_STRIDE + m];
        b.y = hbuf[wave][(ka + 1) * LDS_STRIDE + m];
        outf = __builtin_amdgcn_wmma_f32_16x16x4_f32(
            false, a, false, b, (short)0, outf, false, false);
    }

    float s = 0.0f;
#pragma unroll
    for (int v = 0; v < 8; ++v) s += outf[v] * outf[v];

    // wave32 butterfly reduce
    for (int off = 16; off > 0; off >>= 1)
        s += __shfl_xor(s, off, 32);

    if (lane == 0) wsum[wave] = s;
    __syncthreads();
    if (threadIdx.x == 0) {
        float t = wsum[0] + wsum[1] + wsum[2] + wsum[3];
        partial[blockIdx.x] = t * (1.0f / 4194304.0f);  // mean over N*H*W
    }
}

// Pass 3: deterministic final reduction of block partials -> scalar loss.
__global__ __launch_bounds__(256) void reduce_kernel(const float* __restrict__ partial,
                                                     float* __restrict__ out) {
    __shared__ float buf[256];
    float s = 0.0f;
    for (int i = threadIdx.x; i < NBLK2; i += 256) s += partial[i];
    buf[threadIdx.x] = s;
    __syncthreads();
    for (int off = 128; off > 0; off >>= 1) {
        if ((int)threadIdx.x < off) buf[threadIdx.x] += buf[threadIdx.x + off];
        __syncthreads();
    }
    if (threadIdx.x == 0) out[0] = buf[0];
}

extern "C" void kernel_launch(void* const* d_in, const int* in_sizes, int n_in,
                              void* d_out, int out_size, void* d_ws, size_t ws_size,
                              hipStream_t stream) {
    const float* inp = (const float*)d_in[0];
    const float* tgt = (const float*)d_in[1];
    float* labdiff = (float*)d_ws;                           // 48 * 278784 floats
    float* partial = labdiff + (size_t)NPLANES * PPLANE;     // + 12288 floats

    zeropad_kernel<<<(NBORDER + 255) / 256, 256, 0, stream>>>(labdiff);
    labdiff_kernel<<<NPIX / 256, 256, 0, stream>>>(inp, tgt, labdiff);
    boxwmma_kernel<<<NBLK2, 128, 0, stream>>>(labdiff, partial);
    reduce_kernel<<<1, 256, 0, stream>>>(partial, (float*)d_out);
}